// GCNModel_37598143709432
// MI455X (gfx1250) — compile-verified
//
#include <hip/hip_runtime.h>

#define N_NODES 50000
#define N_EDGES 800000
#define NUM_GRAPHS 256
#define HID 64

typedef float v2f __attribute__((ext_vector_type(2)));
typedef float v8f __attribute__((ext_vector_type(8)));

// ---------------------------------------------------------------------------
// zero buffer
__global__ void zero_kernel(float* __restrict__ p, int n) {
    int i = blockIdx.x * blockDim.x + threadIdx.x;
    if (i < n) p[i] = 0.0f;
}

// degree accumulate over dst
__global__ void deg_kernel(const int* __restrict__ dst, float* __restrict__ deg, int nE) {
    int e = blockIdx.x * blockDim.x + threadIdx.x;
    if (e < nE) atomicAdd(&deg[dst[e]], 1.0f);
}

// deg -> deg_inv_sqrt (with self-loop +1)
__global__ void dis_kernel(float* __restrict__ deg, int n) {
    int i = blockIdx.x * blockDim.x + threadIdx.x;
    if (i < n) deg[i] = rsqrtf(deg[i] + 1.0f);
}

// ---------------------------------------------------------------------------
// Layer 1 GEMM: xw = x[N,2] @ W1[2,64]; agg = xw*dis^2 + b1  (K=2, scalar)
__global__ void gemm_in_kernel(const float* __restrict__ x, const float* __restrict__ W1,
                               const float* __restrict__ b1, const float* __restrict__ dis,
                               float* __restrict__ xw, float* __restrict__ agg, int n) {
    int idx = blockIdx.x * blockDim.x + threadIdx.x;
    if (idx >= n * HID) return;
    int node = idx >> 6, c = idx & 63;
    float v = x[node * 2 + 0] * W1[c] + x[node * 2 + 1] * W1[HID + c];
    float d = dis[node];
    xw[idx] = v;
    agg[idx] = v * d * d + b1[c];
}

// ---------------------------------------------------------------------------
// WMMA GEMM: In[N,64] (ReLU applied on load) @ W[64,64] -> xw[N,64]
// also writes agg = xw*dis^2 + b (self-loop + bias pre-accumulated).
// One wave per 16-row tile; V_WMMA_F32_16X16X4_F32, K stepped by 4,
// 4 column tiles of 16 -> full 16x64 output per wave.
__global__ __launch_bounds__(256)
void gemm64_wmma_kernel(const float* __restrict__ In, const float* __restrict__ W,
                        const float* __restrict__ b, const float* __restrict__ dis,
                        float* __restrict__ xw, float* __restrict__ agg, int nblocks16) {
    int wave = (blockIdx.x * blockDim.x + threadIdx.x) >> 5;
    if (wave >= nblocks16) return;               // wave-uniform: EXEC all-ones inside
    int lane  = threadIdx.x & 31;
    int row0  = wave << 4;
    int mrow  = lane & 15;                       // M row (A) / N col (B,C,D) within tile
    int khalf = (lane >> 4) << 1;                // 0 or 2: K sub-pair held by this lane

    const float* arow = In + (row0 + mrow) * HID;

    v8f acc0 = {}, acc1 = {}, acc2 = {}, acc3 = {};
    for (int k0 = 0; k0 < HID; k0 += 4) {
        // A 16x4 layout: lane<16 -> K=k0,k0+1 ; lane>=16 -> K=k0+2,k0+3
        v2f a;
        a.x = fmaxf(arow[k0 + khalf + 0], 0.0f); // fused ReLU of previous layer
        a.y = fmaxf(arow[k0 + khalf + 1], 0.0f);
        const float* wr0 = W + (k0 + khalf + 0) * HID;
        const float* wr1 = W + (k0 + khalf + 1) * HID;
        // B 4x16 layout mirrors A: lane group selects K half, lane&15 selects N
        v2f b0;  b0.x  = wr0[ 0 + mrow]; b0.y  = wr1[ 0 + mrow];
        v2f b1v; b1v.x = wr0[16 + mrow]; b1v.y = wr1[16 + mrow];
        v2f b2v; b2v.x = wr0[32 + mrow]; b2v.y = wr1[32 + mrow];
        v2f b3v; b3v.x = wr0[48 + mrow]; b3v.y = wr1[48 + mrow];
        acc0 = __builtin_amdgcn_wmma_f32_16x16x4_f32(false, a, false, b0,  (short)0, acc0, false, false);
        acc1 = __builtin_amdgcn_wmma_f32_16x16x4_f32(false, a, false, b1v, (short)0, acc1, false, false);
        acc2 = __builtin_amdgcn_wmma_f32_16x16x4_f32(false, a, false, b2v, (short)0, acc2, false, false);
        acc3 = __builtin_amdgcn_wmma_f32_16x16x4_f32(false, a, false, b3v, (short)0, acc3, false, false);
    }

    // C/D layout: VGPR v -> M = v (lanes 0-15) or v+8 (lanes 16-31); N = lane&15
    int mbase = (lane >> 4) << 3;
    int ncol  = lane & 15;
    for (int v = 0; v < 8; ++v) {
        int row = row0 + mbase + v;
        float d = dis[row];
        float d2 = d * d;
        float vals[4] = {acc0[v], acc1[v], acc2[v], acc3[v]};
        #pragma unroll
        for (int nt = 0; nt < 4; ++nt) {
            int col = nt * 16 + ncol;
            float val = vals[nt];
            xw[row * HID + col]  = val;
            agg[row * HID + col] = val * d2 + b[col];
        }
    }
}

// ---------------------------------------------------------------------------
// Edge scatter: agg[dst] += xw[src] * dis[src]*dis[dst]
// 16 consecutive threads cover one edge's 64 channels (float4 chunks) -> coalesced
__global__ void scatter_kernel(const int* __restrict__ src, const int* __restrict__ dst,
                               const float* __restrict__ dis, const float* __restrict__ xw,
                               float* __restrict__ agg, int nE) {
    int t = blockIdx.x * blockDim.x + threadIdx.x;
    if (t >= nE * 16) return;
    int e = t >> 4;
    int c = (t & 15) << 2;
    int s = src[e], d = dst[e];
    float norm = dis[s] * dis[d];
    const float4 v = *(const float4*)(xw + s * HID + c);
    float* p = agg + d * HID + c;
    atomicAdd(p + 0, v.x * norm);
    atomicAdd(p + 1, v.y * norm);
    atomicAdd(p + 2, v.z * norm);
    atomicAdd(p + 3, v.w * norm);
}

// ---------------------------------------------------------------------------
// Global mean-pool (sum + count); ReLU of final layer fused here
__global__ void pool_kernel(const float* __restrict__ agg, const int* __restrict__ batch,
                            float* __restrict__ sums, float* __restrict__ counts, int n) {
    int t = blockIdx.x * blockDim.x + threadIdx.x;
    if (t >= n * 16) return;
    int node = t >> 4;
    int c = (t & 15) << 2;
    int g = batch[node];
    const float4 v = *(const float4*)(agg + node * HID + c);
    float* p = sums + g * HID + c;
    atomicAdd(p + 0, fmaxf(v.x, 0.0f));
    atomicAdd(p + 1, fmaxf(v.y, 0.0f));
    atomicAdd(p + 2, fmaxf(v.z, 0.0f));
    atomicAdd(p + 3, fmaxf(v.w, 0.0f));
    if (c == 0) atomicAdd(&counts[g], 1.0f);
}

// Head: out[g,t] = (sums[g,:]/max(cnt,1)) . Wc[:,t] + bc[t]
__global__ void head_kernel(const float* __restrict__ sums, const float* __restrict__ counts,
                            const float* __restrict__ Wc, const float* __restrict__ bc,
                            float* __restrict__ out) {
    int t = blockIdx.x * blockDim.x + threadIdx.x;
    if (t >= NUM_GRAPHS * 2) return;
    int g = t >> 1, task = t & 1;
    float cnt = fmaxf(counts[g], 1.0f);
    float acc = 0.0f;
    #pragma unroll
    for (int k = 0; k < HID; ++k) acc += sums[g * HID + k] * Wc[k * 2 + task];
    out[t] = acc / cnt + bc[task];
}

// ---------------------------------------------------------------------------
extern "C" void kernel_launch(void* const* d_in, const int* in_sizes, int n_in,
                              void* d_out, int out_size, void* d_ws, size_t ws_size,
                              hipStream_t stream) {
    (void)in_sizes; (void)n_in; (void)out_size; (void)ws_size;
    const float* x     = (const float*)d_in[0];
    const int*   eidx  = (const int*)d_in[1];
    const int*   batch = (const int*)d_in[2];
    const float* W1    = (const float*)d_in[3];
    const float* b1    = (const float*)d_in[4];
    const float* W2    = (const float*)d_in[5];
    const float* b2    = (const float*)d_in[6];
    const float* W3    = (const float*)d_in[7];
    const float* b3    = (const float*)d_in[8];
    const float* Wc    = (const float*)d_in[9];
    const float* bc    = (const float*)d_in[10];
    float* out = (float*)d_out;

    const int*   src = eidx;            // edge_index[0]
    const int*   dst = eidx + N_EDGES;  // edge_index[1]

    // workspace layout (floats)
    float* ws    = (float*)d_ws;
    float* dis   = ws;                          // N
    float* xw    = dis  + N_NODES;              // N*64
    float* aggA  = xw   + N_NODES * HID;        // N*64
    float* aggB  = aggA + N_NODES * HID;        // N*64
    float* sums  = aggB + N_NODES * HID;        // 256*64
    float* cnts  = sums + NUM_GRAPHS * HID;     // 256 (contiguous with sums)

    const int B = 256;
    const int nb16 = N_NODES / 16;              // 3125 exact

    // zero degree + pool accumulators
    zero_kernel<<<(N_NODES + B - 1) / B, B, 0, stream>>>(dis, N_NODES);
    zero_kernel<<<(NUM_GRAPHS * HID + NUM_GRAPHS + B - 1) / B, B, 0, stream>>>(sums, NUM_GRAPHS * HID + NUM_GRAPHS);

    // degrees -> deg_inv_sqrt
    deg_kernel<<<(N_EDGES + B - 1) / B, B, 0, stream>>>(dst, dis, N_EDGES);
    dis_kernel<<<(N_NODES + B - 1) / B, B, 0, stream>>>(dis, N_NODES);

    // Layer 1 (K=2 scalar GEMM) + scatter
    gemm_in_kernel<<<(N_NODES * HID + B - 1) / B, B, 0, stream>>>(x, W1, b1, dis, xw, aggA, N_NODES);
    scatter_kernel<<<(N_EDGES * 16 + B - 1) / B, B, 0, stream>>>(src, dst, dis, xw, aggA, N_EDGES);

    // Layer 2 (WMMA f32) + scatter   (ReLU of aggA fused into A-load)
    gemm64_wmma_kernel<<<(nb16 * 32 + B - 1) / B, B, 0, stream>>>(aggA, W2, b2, dis, xw, aggB, nb16);
    scatter_kernel<<<(N_EDGES * 16 + B - 1) / B, B, 0, stream>>>(src, dst, dis, xw, aggB, N_EDGES);

    // Layer 3 (WMMA f32) + scatter
    gemm64_wmma_kernel<<<(nb16 * 32 + B - 1) / B, B, 0, stream>>>(aggB, W3, b3, dis, xw, aggA, nb16);
    scatter_kernel<<<(N_EDGES * 16 + B - 1) / B, B, 0, stream>>>(src, dst, dis, xw, aggA, N_EDGES);

    // mean pool (final ReLU fused) + head
    pool_kernel<<<(N_NODES * 16 + B - 1) / B, B, 0, stream>>>(aggA, batch, sums, cnts, N_NODES);
    head_kernel<<<(NUM_GRAPHS * 2 + B - 1) / B, B, 0, stream>>>(sums, cnts, Wc, bc, out);
}